// BEVFeatureExtractor_45784351375627
// MI455X (gfx1250) — compile-verified
//
#include <hip/hip_runtime.h>

// ---- zero-risk feature probes (report via stderr, no codegen impact) ----
#if __has_builtin(__builtin_amdgcn_global_load_async_to_lds_b128)
#warning "PROBE: HAS __builtin_amdgcn_global_load_async_to_lds_b128"
#endif
#if __has_builtin(__builtin_amdgcn_global_load_async_to_lds_b32)
#warning "PROBE: HAS __builtin_amdgcn_global_load_async_to_lds_b32"
#endif
#if __has_builtin(__builtin_amdgcn_cluster_load_async_to_lds_b128)
#warning "PROBE: HAS __builtin_amdgcn_cluster_load_async_to_lds_b128"
#endif
#if __has_builtin(__builtin_amdgcn_load_to_lds)
#warning "PROBE: HAS __builtin_amdgcn_load_to_lds"
#endif
#if __has_builtin(__builtin_amdgcn_s_wait_asynccnt)
#warning "PROBE: HAS __builtin_amdgcn_s_wait_asynccnt"
#endif
#if __has_builtin(__builtin_amdgcn_ds_atomic_async_barrier_arrive_b64)
#warning "PROBE: HAS __builtin_amdgcn_ds_atomic_async_barrier_arrive_b64"
#endif

#define BATCH     2
#define BEV_H     400
#define BEV_W     400
#define C_IN      128
#define C_OUT     256
#define N_PTS     120000
#define PIX_PER_B (BEV_H * BEV_W)        // 160000
#define M_TOTAL   (BATCH * PIX_PER_B)    // 320000
#define BN_EPS    1e-5f

#define LDA  130   // padded LDS row stride (floats) for A/B tiles
#define TPAD 17    // padded stride for transpose buffer

typedef float v2f __attribute__((ext_vector_type(2)));
typedef float v8f __attribute__((ext_vector_type(8)));

// ---------------- Kernel 1: zero grid + BN accumulators ----------------
__global__ void k_zero(float4* __restrict__ p, int n4) {
    int i = blockIdx.x * blockDim.x + threadIdx.x;
    if (i < n4) p[i] = make_float4(0.f, 0.f, 0.f, 0.f);
}

// ---------------- Kernel 2: scatter-max into BEV grid ----------------
// features > 0 and grid zero-initialized, so signed-int atomicMax on the
// float bit pattern is exact (positive floats are monotone as ints).
__global__ void k_scatter(const float* __restrict__ feats,
                          const int*   __restrict__ coords,
                          int*         __restrict__ grid) {
    int t = blockIdx.x * blockDim.x + threadIdx.x;
    if (t >= N_PTS * C_IN) return;
    int pt = t >> 7;         // /128
    int b = coords[pt * 4 + 0];
    int y = coords[pt * 4 + 2];
    int x = coords[pt * 4 + 3];
    float v = feats[t];      // == feats[pt*C_IN + (t&127)]
    int p = b * PIX_PER_B + y * BEV_W + x;
    atomicMax(&grid[p * C_IN + (t & (C_IN - 1))], __float_as_int(v));
}

// Shared helper: cooperative LDS fill of A (64x128) and B (32x128) tiles.
__device__ __forceinline__ void fill_tiles(const float* __restrict__ grid,
                                           const float* __restrict__ Wmat,
                                           float* As, float* Bs,
                                           int m0, int o0, int tid) {
    const float4* src = (const float4*)(grid + (size_t)m0 * C_IN);
    for (int i = tid; i < 64 * 32; i += 256) {       // 64 rows x 32 float4
        int r = i >> 5, c4 = i & 31;
        float4 v = src[r * 32 + c4];
        float* d = &As[r * LDA + c4 * 4];
        d[0] = v.x; d[1] = v.y; d[2] = v.z; d[3] = v.w;
    }
    const float4* wsrc = (const float4*)(Wmat + (size_t)o0 * C_IN);
    for (int i = tid; i < 32 * 32; i += 256) {       // 32 rows x 32 float4
        int r = i >> 5, c4 = i & 31;
        float4 v = wsrc[r * 32 + c4];
        float* d = &Bs[r * LDA + c4 * 4];
        d[0] = v.x; d[1] = v.y; d[2] = v.z; d[3] = v.w;
    }
}

// Shared helper: 16x16x128 f32 WMMA accumulate from padded LDS tiles.
__device__ __forceinline__ v8f wmma_tile(const float* aRow, const float* bRow) {
    v8f acc = {};
    #pragma unroll
    for (int k0 = 0; k0 < C_IN; k0 += 4) {
        v2f a; a.x = aRow[k0]; a.y = aRow[k0 + 1];
        v2f b; b.x = bRow[k0]; b.y = bRow[k0 + 1];
        acc = __builtin_amdgcn_wmma_f32_16x16x4_f32(
            false, a, false, b, (short)0, acc, false, false);
    }
    return acc;
}

// ---------------- Kernel 3 (pass 1): WMMA GEMM -> per-channel sum/sumsq only ----------------
// Block: 256 threads = 8 waves; wave -> m-tile (w&3), n-tile (w>>2).
// Block tile: 64 pixels x 32 channels, K = 128. No output store.
__global__ __launch_bounds__(256) void k_gemm_stats(const float* __restrict__ grid,
                                                    const float* __restrict__ Wmat,
                                                    float* __restrict__ gSum,
                                                    float* __restrict__ gSq) {
    __shared__ float As[64 * LDA];
    __shared__ float Bs[32 * LDA];
    __shared__ float sSum[32], sSq[32];

    const int tid  = threadIdx.x;
    const int wave = tid >> 5;
    const int lane = tid & 31;
    const int half = lane >> 4;
    const int l16  = lane & 15;
    const int mt   = wave & 3;
    const int nt   = wave >> 2;

    const int m0 = blockIdx.x * 64;
    const int o0 = blockIdx.y * 32;

    if (tid < 32) { sSum[tid] = 0.f; sSq[tid] = 0.f; }
    fill_tiles(grid, Wmat, As, Bs, m0, o0, tid);
    __syncthreads();

    v8f acc = wmma_tile(&As[(mt * 16 + l16) * LDA + half * 2],
                        &Bs[(nt * 16 + l16) * LDA + half * 2]);

    // acc[r] = element (M = r + half*8, N = l16): channel is lane-resident.
    float s = 0.f, q = 0.f;
    #pragma unroll
    for (int r = 0; r < 8; ++r) { float v = acc[r]; s += v; q += v * v; }
    s += __shfl_xor(s, 16);  q += __shfl_xor(q, 16);   // fold M halves
    if (half == 0) {
        atomicAdd(&sSum[nt * 16 + l16], s);
        atomicAdd(&sSq [nt * 16 + l16], q);
    }
    __syncthreads();
    if (tid < 32)       atomicAdd(&gSum[o0 + tid], sSum[tid]);
    else if (tid < 64)  atomicAdd(&gSq [o0 + tid - 32], sSq[tid - 32]);
}

// ---------------- Kernel 4 (pass 2): WMMA GEMM + fused BN + ReLU + NT store ----------------
__global__ __launch_bounds__(256) void k_gemm_norm(const float* __restrict__ grid,
                                                   const float* __restrict__ Wmat,
                                                   const float* __restrict__ gSum,
                                                   const float* __restrict__ gSq,
                                                   const float* __restrict__ gamma,
                                                   const float* __restrict__ beta,
                                                   float* __restrict__ out) {
    __shared__ float As[64 * LDA];
    __shared__ float Bs[32 * LDA];
    __shared__ float T[8 * 16 * TPAD];
    __shared__ float ssc[32], ssh[32];

    const int tid  = threadIdx.x;
    const int wave = tid >> 5;
    const int lane = tid & 31;
    const int half = lane >> 4;
    const int l16  = lane & 15;
    const int mt   = wave & 3;
    const int nt   = wave >> 2;

    const int m0 = blockIdx.x * 64;
    const int o0 = blockIdx.y * 32;

    // Per-block BN scale/shift for this 32-channel slab.
    if (tid < 32) {
        const float inv = 1.0f / (float)M_TOTAL;
        int o = o0 + tid;
        float mean = gSum[o] * inv;
        float var  = gSq[o] * inv - mean * mean;
        float sc   = gamma[o] * rsqrtf(var + BN_EPS);
        ssc[tid] = sc;
        ssh[tid] = beta[o] - mean * sc;
    }
    fill_tiles(grid, Wmat, As, Bs, m0, o0, tid);
    __syncthreads();

    v8f acc = wmma_tile(&As[(mt * 16 + l16) * LDA + half * 2],
                        &Bs[(nt * 16 + l16) * LDA + half * 2]);

    // Fuse BN + ReLU (channel = l16 within wave's n-tile).
    {
        float sc = ssc[nt * 16 + l16];
        float sh = ssh[nt * 16 + l16];
        #pragma unroll
        for (int r = 0; r < 8; ++r)
            acc[r] = fmaxf(fmaf(acc[r], sc, sh), 0.f);
    }

    // Transpose via padded LDS, then coalesced non-temporal NCHW store.
    float* Tw = &T[wave * 16 * TPAD];
    #pragma unroll
    for (int r = 0; r < 8; ++r)
        Tw[(r + half * 8) * TPAD + l16] = acc[r];
    __syncthreads();

    const int pBase = m0 + mt * 16;               // 64 | 160000: never straddles batch
    const int b     = pBase / PIX_PER_B;
    const int rem   = pBase - b * PIX_PER_B;
    float* outB = out + (size_t)b * C_OUT * PIX_PER_B + rem + l16;

    #pragma unroll
    for (int i = 0; i < 8; ++i) {
        const int cl = i * 2 + half;
        float v = Tw[l16 * TPAD + cl];
        __builtin_nontemporal_store(v, &outB[(size_t)(o0 + nt * 16 + cl) * PIX_PER_B]);
    }
}

extern "C" void kernel_launch(void* const* d_in, const int* in_sizes, int n_in,
                              void* d_out, int out_size, void* d_ws, size_t ws_size,
                              hipStream_t stream) {
    const float* features = (const float*)d_in[0];
    const int*   coords   = (const int*)  d_in[1];
    const float* conv_w   = (const float*)d_in[2];
    const float* gamma    = (const float*)d_in[3];
    const float* beta     = (const float*)d_in[4];
    float* out = (float*)d_out;

    const int GRID_FLOATS = BATCH * BEV_H * BEV_W * C_IN;   // 40,960,000
    float* grid = (float*)d_ws;
    float* gSum = grid + GRID_FLOATS;
    float* gSq  = gSum + C_OUT;

    // 1) zero grid + accumulators
    int n4 = (GRID_FLOATS + 2 * C_OUT) / 4;
    k_zero<<<(n4 + 255) / 256, 256, 0, stream>>>((float4*)d_ws, n4);

    // 2) scatter-max
    int nscat = N_PTS * C_IN;
    k_scatter<<<(nscat + 255) / 256, 256, 0, stream>>>(features, coords, (int*)grid);

    // 3) pass 1: stats-only WMMA GEMM
    dim3 g3(M_TOTAL / 64, C_OUT / 32);
    k_gemm_stats<<<g3, 256, 0, stream>>>(grid, conv_w, gSum, gSq);

    // 4) pass 2: WMMA GEMM + fused BN/ReLU, non-temporal output store
    k_gemm_norm<<<g3, 256, 0, stream>>>(grid, conv_w, gSum, gSq, gamma, beta, out);
}